// DBNSigma_Single_17987323036449
// MI455X (gfx1250) — compile-verified
//
#include <hip/hip_runtime.h>

// ---------------------------------------------------------------------------
// ZCA whitening (DBN) for X[128,64,56,56] f32 on gfx1250.
//   sigma = eps*I + cov(channels);  wm = sigma^{-1/2} (Newton-Schulz);
//   out = wm @ (x - mean)  ==  wm@x - (wm@mean)
// Big GEMMs: V_WMMA_F32_16X16X4_F32 on LDS tiles.
// Global->LDS staging: GLOBAL_LOAD_ASYNC_TO_LDS_B128 (ASYNCcnt), double-buffered.
// ---------------------------------------------------------------------------

typedef __attribute__((ext_vector_type(2))) float v2f;
typedef __attribute__((ext_vector_type(8))) float v8f;

#define CCH   64                 // channels
#define TS    64                 // positions per LDS tile
#define LSTR  68                 // padded LDS row stride (floats): bank-conflict-free
#define HW    3136               // 56*56
#define NIMG  128
#define MTOT  (NIMG * HW)        // 401408 positions
#define CPI   (HW / TS)          // 49 chunks per image
#define NCHUNK (MTOT / TS)       // 6272
#define ZCA_EPS 1e-3f
#define NS_ITERS 12

// ---- CDNA5 async global->LDS copy (ASYNCcnt) ------------------------------
__device__ __forceinline__ void async_ld_b128(unsigned lds_byte_off,
                                              const float* gaddr) {
  // GV mode: copies 16 bytes per lane from global vaddr to LDS offset vdst, tracked by ASYNCcnt
  asm volatile("global_load_async_to_lds_b128 %0, %1, off"
               :: "v"(lds_byte_off), "v"(gaddr) : "memory");
}
__device__ __forceinline__ void wait_async0() {
  asm volatile("s_wait_asynccnt 0x0" ::: "memory");
}
__device__ __forceinline__ unsigned lds_off(const void* p) {
  // generic LDS pointer: low 32 bits are the workgroup-relative LDS address
  return (unsigned)(unsigned long long)p;
}

// ---- WMMA tile helpers (ISA 7.12.2 f32 layouts) ---------------------------
// A (16x4, MxK): lane L<16: M=L, holds K=0(vgpr0),K=1(vgpr1); lane 16+L: K=2,3
// B (4x16, KxN): lane L<16: N=L, holds K=0,1; lane 16+L: K=2,3
// C/D (16x16):   vgpr r: lanes 0-15 -> (M=r, N=lane); lanes 16-31 -> (M=r+8)

__device__ __forceinline__ v8f mm_tile_64(const float* __restrict__ P,
                                          const float* __restrict__ Q,
                                          int bi, int bj, int mrow, int kb) {
  v8f acc = {};
  #pragma unroll
  for (int k0 = 0; k0 < CCH; k0 += 4) {
    v2f a, b;
    a.x = P[(bi * 16 + mrow) * LSTR + k0 + kb];
    a.y = P[(bi * 16 + mrow) * LSTR + k0 + kb + 1];
    b.x = Q[(k0 + kb) * LSTR + bj * 16 + mrow];
    b.y = Q[(k0 + kb + 1) * LSTR + bj * 16 + mrow];
    acc = __builtin_amdgcn_wmma_f32_16x16x4_f32(false, a, false, b,
                                                (short)0, acc, false, false);
  }
  return acc;
}

__device__ __forceinline__ void store_tile_lds(float* __restrict__ D, v8f acc,
                                               int bi, int bj, int mrow, int hi) {
  const int col = bj * 16 + mrow;
  #pragma unroll
  for (int r = 0; r < 8; ++r)
    D[(bi * 16 + r + hi) * LSTR + col] = acc[r];
}

// ---------------------------------------------------------------------------
__global__ void zero_kernel(float* __restrict__ p, int n) {
  int i = blockIdx.x * blockDim.x + threadIdx.x;
  if (i < n) p[i] = 0.0f;
}

// ---- Pass 1: raw second moment S = x x^T (64x64) and channel sums ---------
__global__ __launch_bounds__(512) void cov_kernel(const float* __restrict__ X,
                                                  float* __restrict__ S,
                                                  float* __restrict__ sums) {
  __shared__ float T[2][CCH * LSTR];
  const int t    = threadIdx.x;
  const int lane = t & 31;
  const int wave = t >> 5;           // 16 waves
  const int bi   = wave >> 2, bj = wave & 3;
  const int mrow = lane & 15;
  const int kb   = (lane < 16) ? 0 : 2;
  const int hi   = (lane < 16) ? 0 : 8;

  // this thread's two float4 slots within a 64x64 tile
  const int ch0 = t >> 4;              // f = t
  const int jb0 = (t & 15) * 4;
  const int ch1 = (t + 512) >> 4;      // f = t + 512
  const int jb1 = ((t + 512) & 15) * 4;

  v8f acc = {};
  float csum = 0.0f;

  const int q0 = blockIdx.x;           // gridDim <= NCHUNK: every block has work
  // prefetch first tile into buffer 0
  {
    const int n   = q0 / CPI;
    const int hw0 = (q0 - n * CPI) * TS;
    const float* base = X + (size_t)n * (CCH * HW) + hw0;
    async_ld_b128(lds_off(&T[0][ch0 * LSTR + jb0]), base + (size_t)ch0 * HW + jb0);
    async_ld_b128(lds_off(&T[0][ch1 * LSTR + jb1]), base + (size_t)ch1 * HW + jb1);
  }

  int buf = 0;
  for (int q = q0; q < NCHUNK; q += gridDim.x) {
    wait_async0();        // my async writes into T[buf] landed
    __syncthreads();      // everyone's landed; everyone done reading T[buf^1]

    const int qn = q + gridDim.x;
    if (qn < NCHUNK) {    // block-uniform: prefetch next tile into T[buf^1]
      const int n   = qn / CPI;
      const int hw0 = (qn - n * CPI) * TS;
      const float* base = X + (size_t)n * (CCH * HW) + hw0;
      async_ld_b128(lds_off(&T[buf ^ 1][ch0 * LSTR + jb0]), base + (size_t)ch0 * HW + jb0);
      async_ld_b128(lds_off(&T[buf ^ 1][ch1 * LSTR + jb1]), base + (size_t)ch1 * HW + jb1);
    }

    const float* Tb = &T[buf][0];
    if (t < CCH) {                   // fold channel-sum reduction in
      float s = 0.0f;
      #pragma unroll 8
      for (int j = 0; j < TS; ++j) s += Tb[t * LSTR + j];
      csum += s;
    }

    const int ca = bi * 16 + mrow;   // A rows (channels)
    const int cb = bj * 16 + mrow;   // B cols (channels); K = position
    #pragma unroll
    for (int k0 = 0; k0 < TS; k0 += 4) {
      v2f a, b;
      a.x = Tb[ca * LSTR + k0 + kb];
      a.y = Tb[ca * LSTR + k0 + kb + 1];
      b.x = Tb[cb * LSTR + k0 + kb];
      b.y = Tb[cb * LSTR + k0 + kb + 1];
      acc = __builtin_amdgcn_wmma_f32_16x16x4_f32(false, a, false, b,
                                                  (short)0, acc, false, false);
    }
    buf ^= 1;
  }

  const int col = bj * 16 + (lane & 15);
  #pragma unroll
  for (int r = 0; r < 8; ++r) {
    const int row = bi * 16 + r + hi;
    atomicAdd(&S[row * CCH + col], acc[r]);
  }
  if (t < CCH) atomicAdd(&sums[t], csum);
}

// ---- Pass 2: single-block solve: sigma -> sigma^{-1/2} via Newton-Schulz --
__global__ __launch_bounds__(512) void solve_kernel(const float* __restrict__ S,
                                                    const float* __restrict__ sums,
                                                    float* __restrict__ wm_out,
                                                    float* __restrict__ bias_out) {
  __shared__ float Y[CCH * LSTR];
  __shared__ float Z[CCH * LSTR];
  __shared__ float B1[CCH * LSTR];
  __shared__ float mean[CCH];
  __shared__ float sInvTr, sRsq;

  const int t    = threadIdx.x;
  const int lane = t & 31;
  const int wave = t >> 5;
  const int bi   = wave >> 2, bj = wave & 3;
  const int mrow = lane & 15;
  const int kb   = (lane < 16) ? 0 : 2;
  const int hi   = (lane < 16) ? 0 : 8;

  const float invM = 1.0f / (float)MTOT;
  if (t < CCH) mean[t] = sums[t] * invM;
  __syncthreads();

  // sigma -> B1
  for (int e = t; e < CCH * CCH; e += 512) {
    const int i = e >> 6, j = e & 63;
    B1[i * LSTR + j] = S[e] * invM - mean[i] * mean[j] + ((i == j) ? ZCA_EPS : 0.0f);
  }
  __syncthreads();
  if (t == 0) {
    float tr = 0.0f;
    for (int i = 0; i < CCH; ++i) tr += B1[i * LSTR + i];
    sInvTr = 1.0f / tr;
    sRsq   = rsqrtf(tr);
  }
  __syncthreads();
  // Y0 = sigma/tr (spectrum in (0,1] -> NS converges), Z0 = I
  for (int e = t; e < CCH * CCH; e += 512) {
    const int i = e >> 6, j = e & 63;
    Y[i * LSTR + j] = B1[i * LSTR + j] * sInvTr;
    Z[i * LSTR + j] = (i == j) ? 1.0f : 0.0f;
  }
  __syncthreads();

  for (int it = 0; it < NS_ITERS; ++it) {
    // B1 = Z @ Y
    v8f m = mm_tile_64(Z, Y, bi, bj, mrow, kb);
    __syncthreads();
    store_tile_lds(B1, m, bi, bj, mrow, hi);
    __syncthreads();
    // B1 = 3I - B1
    for (int e = t; e < CCH * CCH; e += 512) {
      const int i = e >> 6, j = e & 63;
      B1[i * LSTR + j] = ((i == j) ? 3.0f : 0.0f) - B1[i * LSTR + j];
    }
    __syncthreads();
    // Y = 0.5 * Y @ B1   (in-place: all reads done before barrier, then write)
    v8f y2 = mm_tile_64(Y, B1, bi, bj, mrow, kb);
    __syncthreads();
    #pragma unroll
    for (int r = 0; r < 8; ++r) y2[r] *= 0.5f;
    store_tile_lds(Y, y2, bi, bj, mrow, hi);
    __syncthreads();
    // Z = 0.5 * B1 @ Z
    v8f z2 = mm_tile_64(B1, Z, bi, bj, mrow, kb);
    __syncthreads();
    #pragma unroll
    for (int r = 0; r < 8; ++r) z2[r] *= 0.5f;
    store_tile_lds(Z, z2, bi, bj, mrow, hi);
    __syncthreads();
  }

  // wm = Z * tr^{-1/2};  bias = wm @ mean
  const float rsq = sRsq;
  for (int e = t; e < CCH * CCH; e += 512) {
    const int i = e >> 6, j = e & 63;
    wm_out[e] = Z[i * LSTR + j] * rsq;
  }
  __syncthreads();
  if (t < CCH) {
    float b = 0.0f;
    for (int j = 0; j < CCH; ++j) b += Z[t * LSTR + j] * rsq * mean[j];
    bias_out[t] = b;
  }
}

// ---- Pass 3: out = wm @ x - bias ------------------------------------------
__global__ __launch_bounds__(512) void apply_kernel(const float* __restrict__ X,
                                                    const float* __restrict__ wm,
                                                    const float* __restrict__ bias,
                                                    float* __restrict__ out) {
  __shared__ float W[CCH * LSTR];
  __shared__ float T[2][CCH * LSTR];
  __shared__ float bsh[CCH];

  const int t    = threadIdx.x;
  const int lane = t & 31;
  const int wave = t >> 5;
  const int bi   = wave >> 2, bj = wave & 3;
  const int mrow = lane & 15;
  const int kb   = (lane < 16) ? 0 : 2;
  const int hi   = (lane < 16) ? 0 : 8;
  const int col  = bj * 16 + mrow;

  const int ch0 = t >> 4;
  const int jb0 = (t & 15) * 4;
  const int ch1 = (t + 512) >> 4;
  const int jb1 = ((t + 512) & 15) * 4;

  for (int e = t; e < CCH * CCH; e += 512)
    W[(e >> 6) * LSTR + (e & 63)] = wm[e];
  if (t < CCH) bsh[t] = bias[t];

  const int q0 = blockIdx.x;
  {
    const int n   = q0 / CPI;
    const int hw0 = (q0 - n * CPI) * TS;
    const float* base = X + (size_t)n * (CCH * HW) + hw0;
    async_ld_b128(lds_off(&T[0][ch0 * LSTR + jb0]), base + (size_t)ch0 * HW + jb0);
    async_ld_b128(lds_off(&T[0][ch1 * LSTR + jb1]), base + (size_t)ch1 * HW + jb1);
  }

  int buf = 0;
  for (int q = q0; q < NCHUNK; q += gridDim.x) {
    wait_async0();
    __syncthreads();

    const int qn = q + gridDim.x;
    if (qn < NCHUNK) {
      const int n   = qn / CPI;
      const int hw0 = (qn - n * CPI) * TS;
      const float* base = X + (size_t)n * (CCH * HW) + hw0;
      async_ld_b128(lds_off(&T[buf ^ 1][ch0 * LSTR + jb0]), base + (size_t)ch0 * HW + jb0);
      async_ld_b128(lds_off(&T[buf ^ 1][ch1 * LSTR + jb1]), base + (size_t)ch1 * HW + jb1);
    }

    const float* Tb = &T[buf][0];
    // O(64x64) = W(64x64) @ T(64 ch x 64 pos): M=out ch, K=ch, N=pos
    v8f acc = {};
    #pragma unroll
    for (int k0 = 0; k0 < CCH; k0 += 4) {
      v2f a, b;
      a.x = W[(bi * 16 + mrow) * LSTR + k0 + kb];
      a.y = W[(bi * 16 + mrow) * LSTR + k0 + kb + 1];
      b.x = Tb[(k0 + kb) * LSTR + col];
      b.y = Tb[(k0 + kb + 1) * LSTR + col];
      acc = __builtin_amdgcn_wmma_f32_16x16x4_f32(false, a, false, b,
                                                  (short)0, acc, false, false);
    }

    const int n   = q / CPI;
    const int hw0 = (q - n * CPI) * TS;
    float* obase = out + (size_t)n * (CCH * HW) + hw0;
    #pragma unroll
    for (int r = 0; r < 8; ++r) {
      const int ch = bi * 16 + r + hi;
      obase[(size_t)ch * HW + col] = acc[r] - bsh[ch];
    }
    buf ^= 1;
  }
}

// ---------------------------------------------------------------------------
extern "C" void kernel_launch(void* const* d_in, const int* in_sizes, int n_in,
                              void* d_out, int out_size, void* d_ws, size_t ws_size,
                              hipStream_t stream) {
  (void)in_sizes; (void)n_in; (void)out_size; (void)ws_size;
  const float* X = (const float*)d_in[0];
  float* out = (float*)d_out;
  float* ws  = (float*)d_ws;

  float* S    = ws;                      // 64*64 = 4096 floats
  float* sums = ws + 4096;               // 64
  float* wm   = ws + 4096 + 128;         // 4096 (aligned)
  float* bias = ws + 4096 + 128 + 4096;  // 64

  zero_kernel<<<(4160 + 255) / 256, 256, 0, stream>>>(ws, 4160);
  cov_kernel<<<512, 512, 0, stream>>>(X, S, sums);
  solve_kernel<<<1, 512, 0, stream>>>(S, sums, wm, bias);
  apply_kernel<<<512, 512, 0, stream>>>(X, wm, bias, out);
}